// PrimitiveLibrary_36524401885469
// MI455X (gfx1250) — compile-verified
//
#include <hip/hip_runtime.h>
#include <math.h>

typedef float v2f __attribute__((ext_vector_type(2)));
typedef float v8f __attribute__((ext_vector_type(8)));

#define SD 4096
#define SPLIT_K 8
#define WAVES_PER_BLOCK 8
#define ROWS_PER_BLOCK (16 * WAVES_PER_BLOCK)   // 128 rows per block
#define GEMV_BLOCK 256

#define ACT_NONE 0
#define ACT_TANH 1
#define ACT_GELU 2

__device__ __forceinline__ float gelu_erf(float x) {
    return 0.5f * x * (1.0f + erff(x * 0.70710678118654752f));
}
__device__ __forceinline__ float sigmoid_f(float x) {
    return 1.0f / (1.0f + expf(-x));
}

// ---------------------------------------------------------------------------
// GEMV via V_WMMA_F32_16X16X4_F32.
// Each wave computes 16 output rows over this block's K-chunk (in_dim/SPLIT_K).
// A layout (ISA 7.12.2, f32 16x4): lane L<16 holds row L (K0,K1) in the VGPR
// pair, lane L+16 holds row L (K2,K3) -> load directly as one b64 per lane.
// B = x replicated across all 16 N columns (same K0|K2 / K1|K3 half-split),
// so every D column holds y; lanes 0/16 write the 16 row results.
// Partial sums written to part[blockIdx.y * out_dim + row].
// ---------------------------------------------------------------------------
__global__ __launch_bounds__(GEMV_BLOCK)
void gemv_wmma_kernel(const float* __restrict__ W, const float* __restrict__ x,
                      float* __restrict__ part, int out_dim, int in_dim)
{
    __shared__ float xs[1024];                    // max K-chunk = 8192/8
    const int klen  = in_dim / SPLIT_K;           // 512 or 1024, multiple of 64
    const int kbase = blockIdx.y * klen;

    for (int i = threadIdx.x; i < klen; i += GEMV_BLOCK)
        xs[i] = x[kbase + i];
    __syncthreads();

    const int  lane = threadIdx.x & 31;
    const int  wave = threadIdx.x >> 5;
    const int  row0 = blockIdx.x * ROWS_PER_BLOCK + wave * 16;
    const int  r    = lane & 15;
    const bool hi   = lane >= 16;
    const int  hofs = hi ? 2 : 0;                 // K-half offset within a tile

    const float* wp = W + (size_t)(row0 + r) * in_dim + kbase + hofs;
    const float* bx = xs + hofs;

    v8f acc = {0.f, 0.f, 0.f, 0.f, 0.f, 0.f, 0.f, 0.f};

    #pragma unroll 8
    for (int kk = 0; kk < klen; kk += 8) {
        // A tiles: direct b64 loads in striped layout (no repacking needed)
        const v2f a0 = *reinterpret_cast<const v2f*>(wp + kk);      // K = kk+0..3
        const v2f a1 = *reinterpret_cast<const v2f*>(wp + kk + 4);  // K = kk+4..7
        // B tiles: x broadcast from LDS in the mirrored layout
        const v2f b0 = *reinterpret_cast<const v2f*>(bx + kk);
        const v2f b1 = *reinterpret_cast<const v2f*>(bx + kk + 4);
        acc = __builtin_amdgcn_wmma_f32_16x16x4_f32(false, a0, false, b0,
                                                    (short)0, acc, false, false);
        acc = __builtin_amdgcn_wmma_f32_16x16x4_f32(false, a1, false, b1,
                                                    (short)0, acc, false, false);
    }

    // D layout: vgpr j -> M=j (lanes 0-15, N=lane), M=j+8 (lanes 16-31).
    // All columns identical; lane 0 writes rows row0..row0+7, lane 16 rows +8..+15.
    if (r == 0) {
        float* po = part + (size_t)blockIdx.y * out_dim + row0 + (hi ? 8 : 0);
        #pragma unroll
        for (int j = 0; j < 8; ++j) po[j] = acc[j];
    }
}

// ---------------------------------------------------------------------------
__global__ void reduce_bias_act(const float* __restrict__ part,
                                const float* __restrict__ bias,
                                float* __restrict__ y, int out_dim, int act)
{
    const int i = blockIdx.x * blockDim.x + threadIdx.x;
    if (i >= out_dim) return;
    float s = bias[i];
    #pragma unroll
    for (int c = 0; c < SPLIT_K; ++c) s += part[(size_t)c * out_dim + i];
    if (act == ACT_TANH)      s = tanhf(s);
    else if (act == ACT_GELU) s = gelu_erf(s);
    y[i] = s;
}

__global__ void concat_kernel(const float* __restrict__ a, const float* __restrict__ b,
                              float* __restrict__ o)
{
    const int i = blockIdx.x * blockDim.x + threadIdx.x;
    if (i < SD) o[i] = a[i];
    else if (i < 2 * SD) o[i] = b[i - SD];
}

// LayerNorm over 4096 elems + affine + erf-GELU, one block of 1024 threads
__global__ __launch_bounds__(1024)
void ln_gelu_kernel(const float* __restrict__ h, const float* __restrict__ g,
                    const float* __restrict__ beta, float* __restrict__ o)
{
    __shared__ float red[1024];
    __shared__ float s_mu, s_rs;
    const int t = threadIdx.x;
    const float v0 = h[t], v1 = h[t + 1024], v2 = h[t + 2048], v3 = h[t + 3072];
    red[t] = v0 + v1 + v2 + v3;
    __syncthreads();
    for (int s = 512; s > 0; s >>= 1) { if (t < s) red[t] += red[t + s]; __syncthreads(); }
    if (t == 0) s_mu = red[0] * (1.0f / 4096.0f);
    __syncthreads();
    const float mu = s_mu;
    const float d0 = v0 - mu, d1 = v1 - mu, d2 = v2 - mu, d3 = v3 - mu;
    red[t] = d0 * d0 + d1 * d1 + d2 * d2 + d3 * d3;
    __syncthreads();
    for (int s = 512; s > 0; s >>= 1) { if (t < s) red[t] += red[t + s]; __syncthreads(); }
    if (t == 0) s_rs = rsqrtf(red[0] * (1.0f / 4096.0f) + 1e-5f);
    __syncthreads();
    const float rs = s_rs;
    o[t]        = gelu_erf(d0 * rs * g[t]        + beta[t]);
    o[t + 1024] = gelu_erf(d1 * rs * g[t + 1024] + beta[t + 1024]);
    o[t + 2048] = gelu_erf(d2 * rs * g[t + 2048] + beta[t + 2048]);
    o[t + 3072] = gelu_erf(d3 * rs * g[t + 3072] + beta[t + 3072]);
}

// sigmoid(dot(q,k)/32) over 1024 elems, one block
__global__ __launch_bounds__(1024)
void dot_sig_kernel(const float* __restrict__ q, const float* __restrict__ k,
                    float* __restrict__ scal)
{
    __shared__ float red[1024];
    const int t = threadIdx.x;
    red[t] = q[t] * k[t];
    __syncthreads();
    for (int s = 512; s > 0; s >>= 1) { if (t < s) red[t] += red[t + s]; __syncthreads(); }
    if (t == 0) scal[0] = sigmoid_f(red[0] * (1.0f / 32.0f));
}

__global__ void scale_kernel(const float* __restrict__ v, const float* __restrict__ scal,
                             float* __restrict__ o)
{
    const int i = blockIdx.x * blockDim.x + threadIdx.x;
    if (i < SD) o[i] = scal[0] * v[i];
}

__global__ void combine_kernel(const float* __restrict__ s,
                               const float* __restrict__ d0, const float* __restrict__ d1,
                               const float* __restrict__ d2, const float* __restrict__ d3,
                               const float* __restrict__ d4, const float* __restrict__ d5,
                               const float* __restrict__ w, const float* __restrict__ gates,
                               float* __restrict__ out)
{
    const int i = blockIdx.x * blockDim.x + threadIdx.x;
    if (i >= SD) return;
    const float* ds[6] = {d0, d1, d2, d3, d4, d5};
    const float si = s[i];
    float acc = 0.0f;
    #pragma unroll
    for (int p = 0; p < 6; ++p) {
        const float gp = sigmoid_f(gates[p]);
        acc += w[p] * (si + gp * ds[p][i]);
    }
    out[i] = acc;
}

// ---------------------------------------------------------------------------
extern "C" void kernel_launch(void* const* d_in, const int* in_sizes, int n_in,
                              void* d_out, int out_size, void* d_ws, size_t ws_size,
                              hipStream_t stream)
{
    (void)in_sizes; (void)n_in; (void)out_size; (void)ws_size;
    const float* weights  = (const float*)d_in[0];
    const float* state    = (const float*)d_in[1];
    const float* id_w     = (const float*)d_in[2];
    const float* id_b     = (const float*)d_in[3];
    const float* neg_w1   = (const float*)d_in[4];
    const float* neg_b1   = (const float*)d_in[5];
    const float* neg_w2   = (const float*)d_in[6];
    const float* neg_b2   = (const float*)d_in[7];
    const float* neg_w3   = (const float*)d_in[8];
    const float* neg_b3   = (const float*)d_in[9];
    const float* mor_w1   = (const float*)d_in[10];
    const float* mor_b1   = (const float*)d_in[11];
    const float* mor_g    = (const float*)d_in[12];
    const float* mor_beta = (const float*)d_in[13];
    const float* mor_w2   = (const float*)d_in[14];
    const float* mor_b2   = (const float*)d_in[15];
    const float* aq_w     = (const float*)d_in[16];
    const float* aq_b     = (const float*)d_in[17];
    const float* ak_w     = (const float*)d_in[18];
    const float* ak_b     = (const float*)d_in[19];
    const float* av_w     = (const float*)d_in[20];
    const float* av_b     = (const float*)d_in[21];
    const float* ao_w     = (const float*)d_in[22];
    const float* ao_b     = (const float*)d_in[23];
    const float* lk_w1    = (const float*)d_in[24];
    const float* lk_b1    = (const float*)d_in[25];
    const float* lk_w2    = (const float*)d_in[26];
    const float* lk_b2    = (const float*)d_in[27];
    const float* lk_w3    = (const float*)d_in[28];
    const float* lk_b3    = (const float*)d_in[29];
    const float* bctx     = (const float*)d_in[30];
    const float* bl_w1    = (const float*)d_in[31];
    const float* bl_b1    = (const float*)d_in[32];
    const float* bl_w2    = (const float*)d_in[33];
    const float* bl_b2    = (const float*)d_in[34];
    const float* gates    = (const float*)d_in[35];
    float* out = (float*)d_out;

    // workspace layout (floats)
    float* ws   = (float*)d_ws;
    float* part = ws;                       // SPLIT_K * 8192 = 65536
    float* d0   = ws + (size_t)SPLIT_K * 8192;
    float* d1   = d0 + SD;
    float* d2   = d1 + SD;
    float* d3   = d2 + SD;
    float* d4   = d3 + SD;
    float* d5   = d4 + SD;
    float* hn1  = d5 + SD;                  // 8192
    float* hl1  = hn1 + 2 * SD;
    float* hn2  = hl1 + 2 * SD;
    float* hl2  = hn2 + 2 * SD;
    float* hm1  = hl2 + 2 * SD;             // 4096
    float* hm   = hm1 + SD;
    float* qv   = hm + SD;                  // 1024
    float* kv   = qv + SD / 4;
    float* vraw = kv + SD / 4;              // 4096
    float* vscl = vraw + SD;
    float* sctx = vscl + SD;                // 8192
    float* hb1  = sctx + 2 * SD;            // 4096
    float* scal = hb1 + SD;                 // 1

    auto GEMV = [&](const float* W, const float* x, const float* b, float* y,
                    int outd, int ind, int act) {
        dim3 grid(outd / ROWS_PER_BLOCK, SPLIT_K);
        gemv_wmma_kernel<<<grid, GEMV_BLOCK, 0, stream>>>(W, x, part, outd, ind);
        reduce_bias_act<<<(outd + 255) / 256, 256, 0, stream>>>(part, b, y, outd, act);
    };

    // stage 0: concat([s, blend_ctx])
    concat_kernel<<<(2 * SD + 255) / 256, 256, 0, stream>>>(state, bctx, sctx);

    // stage 1: everything depending only on s / sctx
    GEMV(id_w,   state, id_b,   d0,  SD,     SD,     ACT_TANH);
    GEMV(neg_w1, state, neg_b1, hn1, 2 * SD, SD,     ACT_GELU);
    GEMV(mor_w1, state, mor_b1, hm1, SD,     SD,     ACT_NONE);
    GEMV(aq_w,   state, aq_b,   qv,  SD / 4, SD,     ACT_NONE);
    GEMV(ak_w,   state, ak_b,   kv,  SD / 4, SD,     ACT_NONE);
    GEMV(av_w,   state, av_b,   vraw,SD,     SD,     ACT_NONE);
    GEMV(lk_w1,  state, lk_b1,  hl1, 2 * SD, SD,     ACT_GELU);
    GEMV(bl_w1,  sctx,  bl_b1,  hb1, SD,     2 * SD, ACT_GELU);

    // stage 2: small fusions + second layers
    ln_gelu_kernel<<<1, 1024, 0, stream>>>(hm1, mor_g, mor_beta, hm);
    dot_sig_kernel<<<1, 1024, 0, stream>>>(qv, kv, scal);
    scale_kernel<<<(SD + 255) / 256, 256, 0, stream>>>(vraw, scal, vscl);
    GEMV(neg_w2, hn1, neg_b2, hn2, 2 * SD, 2 * SD, ACT_GELU);
    GEMV(lk_w2,  hl1, lk_b2,  hl2, 2 * SD, 2 * SD, ACT_GELU);

    // stage 3: output projections
    GEMV(neg_w3, hn2,  neg_b3, d1, SD, 2 * SD, ACT_NONE);
    GEMV(mor_w2, hm,   mor_b2, d2, SD, SD,     ACT_NONE);
    GEMV(ao_w,   vscl, ao_b,   d3, SD, SD,     ACT_NONE);
    GEMV(lk_w3,  hl2,  lk_b3,  d4, SD, 2 * SD, ACT_NONE);
    GEMV(bl_w2,  hb1,  bl_b2,  d5, SD, SD,     ACT_NONE);

    // stage 4: gated residual + soft routing
    combine_kernel<<<(SD + 255) / 256, 256, 0, stream>>>(
        state, d0, d1, d2, d3, d4, d5, weights, gates, out);
}